// BoundaryLoss_25958782337868
// MI455X (gfx1250) — compile-verified
//
#include <hip/hip_runtime.h>
#include <hip/hip_bf16.h>

// Problem constants (from setup_inputs): B=2, C=4, D=96, H=160, W=160
#define BB 2
#define CC 4
#define DD 96
#define HH 160
#define WW 160
#define CHUNKS 8            // row chunks per slice
#define RPC (HH / CHUNKS)   // rows per chunk = 20
#define NBLK (BB * DD * CHUNKS)  // 1536 blocks

typedef __attribute__((ext_vector_type(2))) float v2f;
typedef __attribute__((ext_vector_type(8))) float v8f;

// r_c = softmax_c(pred) - (target == c), or 0 for padded rows
__device__ __forceinline__ void load_r(const float* __restrict__ pb, long cs,
                                       const int* __restrict__ tg,
                                       int h, int w, float r[4]) {
    if (h < 0 || h >= HH) {
        r[0] = r[1] = r[2] = r[3] = 0.0f;
        return;
    }
    long off = (long)h * WW + w;
    float x0 = pb[off];
    float x1 = pb[off + cs];
    float x2 = pb[off + 2 * cs];
    float x3 = pb[off + 3 * cs];
    float m  = fmaxf(fmaxf(x0, x1), fmaxf(x2, x3));
    float e0 = __expf(x0 - m);
    float e1 = __expf(x1 - m);
    float e2 = __expf(x2 - m);
    float e3 = __expf(x3 - m);
    float inv = __builtin_amdgcn_rcpf(e0 + e1 + e2 + e3);
    int t = tg[off];
    r[0] = e0 * inv - (float)(t == 0);
    r[1] = e1 * inv - (float)(t == 1);
    r[2] = e2 * inv - (float)(t == 2);
    r[3] = e3 * inv - (float)(t == 3);
}

// Exact wave32 sum of one float per lane via two stages of
// V_WMMA_F32_16X16X4_F32 with B = all-ones (layout-independent for B).
// Stage 1: A[m][0]=p[m], A[m][2]=p[m+16]  ->  D[m][*] = p[m]+p[m+16]
// Stage 2: each D VGPR r holds s[r] (lanes 0-15) / s[r+8] (lanes 16-31);
// feeding VGPR pairs (d0,d1)..(d6,d7) as A into 4 accumulating WMMAs sums
// all 16 row-sums -> total broadcast into every element of the result.
__device__ __forceinline__ float wave_sum_wmma(float p) {
    v2f a;  a[0] = p;    a[1] = 0.0f;
    v2f bo; bo[0] = 1.0f; bo[1] = 1.0f;
    v8f z = {};
    v8f t1 = __builtin_amdgcn_wmma_f32_16x16x4_f32(false, a, false, bo,
                                                   (short)0, z, false, false);
    v8f c2 = {};
#pragma unroll
    for (int i = 0; i < 4; ++i) {
        v2f a2; a2[0] = t1[2 * i]; a2[1] = t1[2 * i + 1];
        c2 = __builtin_amdgcn_wmma_f32_16x16x4_f32(false, a2, false, bo,
                                                   (short)0, c2, false, false);
    }
    return c2[0];
}

__global__ __launch_bounds__(WW) void boundary_main(
    const float* __restrict__ pred,
    const int* __restrict__ target,
    float* __restrict__ partial) {
    // LDS row buffers: per class, vertical-smooth (s) and vertical-diff (dv),
    // width WW+2 with zero-padded borders for the w +/- 1 taps.
    __shared__ float sL[CC][WW + 2];
    __shared__ float dL[CC][WW + 2];
    __shared__ float wred[5];

    const int w     = threadIdx.x;        // column 0..159
    const int blk   = blockIdx.x;
    const int chunk = blk % CHUNKS;
    const int slice = blk / CHUNKS;       // b*D + d
    const int d     = slice % DD;
    const int b     = slice / DD;

    const long HWl = (long)HH * WW;
    const long cs  = (long)DD * HWl;                      // channel stride
    const float* pb = pred + ((long)b * CC * DD + d) * HWl;  // pred[b,0,d,:,:]
    const int*   tg = target + (long)slice * HWl;            // target[b,d,:,:]

    // zero LDS borders once (never overwritten; w+1 writes land in [1,160])
    if (w < CC) {
        sL[w][0] = 0.0f; sL[w][WW + 1] = 0.0f;
        dL[w][0] = 0.0f; dL[w][WW + 1] = 0.0f;
    }

    const int h0 = chunk * RPC;
    float r0[4], r1[4], r2[4];
    load_r(pb, cs, tg, h0 - 1, w, r0);
    load_r(pb, cs, tg, h0,     w, r1);

    float acc = 0.0f;
    for (int i = 0; i < RPC; ++i) {
        const int h = h0 + i;
        load_r(pb, cs, tg, h + 1, w, r2);
#pragma unroll
        for (int c = 0; c < CC; ++c) {
            sL[c][w + 1] = r0[c] + 2.0f * r1[c] + r2[c];  // [1,2,1] vertical
            dL[c][w + 1] = r2[c] - r0[c];                 // [-1,0,1] vertical
        }
        __syncthreads();
#pragma unroll
        for (int c = 0; c < CC; ++c) {
            float gx = sL[c][w + 2] - sL[c][w];                            // sx
            float gy = dL[c][w] + 2.0f * dL[c][w + 1] + dL[c][w + 2];      // sy==sz
            acc += gx * gx + 2.0f * gy * gy;
        }
        __syncthreads();
#pragma unroll
        for (int c = 0; c < CC; ++c) { r0[c] = r1[c]; r1[c] = r2[c]; }
    }

    // wave32 reduction via f32 WMMA (uniform control flow; EXEC all ones)
    float ws = wave_sum_wmma(acc);
    if ((threadIdx.x & 31) == 0) wred[threadIdx.x >> 5] = ws;
    __syncthreads();
    if (threadIdx.x == 0)
        partial[blk] = wred[0] + wred[1] + wred[2] + wred[3] + wred[4];
}

__global__ __launch_bounds__(256) void boundary_reduce(
    const float* __restrict__ partial, float* __restrict__ out,
    int n, float scale) {
    __shared__ float red[256];
    float s = 0.0f;
    for (int i = threadIdx.x; i < n; i += 256) s += partial[i];
    red[threadIdx.x] = s;
    __syncthreads();
    for (int k = 128; k > 0; k >>= 1) {
        if (threadIdx.x < k) red[threadIdx.x] += red[threadIdx.x + k];
        __syncthreads();
    }
    if (threadIdx.x == 0) out[0] = red[0] * scale;
}

extern "C" void kernel_launch(void* const* d_in, const int* in_sizes, int n_in,
                              void* d_out, int out_size, void* d_ws, size_t ws_size,
                              hipStream_t stream) {
    const float* pred   = (const float*)d_in[0];
    const int*   target = (const int*)d_in[1];
    float* out     = (float*)d_out;
    float* partial = (float*)d_ws;   // NBLK floats = 6 KB

    boundary_main<<<NBLK, WW, 0, stream>>>(pred, target, partial);

    // loss = sum / (B*(D+2)*(H+2)*(W+2)) / C = sum / (2*98*162*162 * 4)
    const float scale = 1.0f / 20575296.0f;
    boundary_reduce<<<1, 256, 0, stream>>>(partial, out, NBLK, scale);
}